// Model_72584947302868
// MI455X (gfx1250) — compile-verified
//
#include <hip/hip_runtime.h>
#include <math.h>

typedef __attribute__((ext_vector_type(2))) float v2f;
typedef __attribute__((ext_vector_type(8))) float v8f;

#define BATCH 32
#define QLEN  512
#define KLEN  511
#define DDIM  768
#define QT    16                 // q rows per workgroup
#define NWAVE 4
#define TPB   (NWAVE * 32)
#define LDS_STRIDE (DDIM + 4)    // 772 floats: stride%64banks==4 -> conflict-free
#define S_STRIDE   (512 + 4)     // 516 floats
#define INV_KEEP   (1.0f / 0.9f)
#define CHUNK_OPS  (QT * (DDIM / 4) / TPB)   // 24 async b128 ops per thread/chunk

#if __has_builtin(__builtin_amdgcn_global_load_async_to_lds_b128)
#define HAVE_ASYNC 1
typedef int v4i_ __attribute__((vector_size(16)));
typedef __attribute__((address_space(1))) v4i_ as1_v4i;
typedef __attribute__((address_space(3))) v4i_ as3_v4i;
#define WAIT_ASYNC_PREV() asm volatile("s_wait_asynccnt 24" ::: "memory")
#define WAIT_ASYNC_ALL()  asm volatile("s_wait_asynccnt 0"  ::: "memory")
#else
#define HAVE_ASYNC 0
#define WAIT_ASYNC_PREV() ((void)0)
#define WAIT_ASYNC_ALL()  ((void)0)
#endif

// Stage 16 rows x 768 f32 from gbase (row-major, DDIM stride) into an LDS tile.
// Rows >= rowLimit are clamped to row 0 (caller zeroes them later if needed).
__device__ __forceinline__ void stage16(const float* __restrict__ gbase,
                                        float* __restrict__ dst,
                                        int row0, int rowLimit, int tid) {
#pragma unroll 4
  for (int i = 0; i < CHUNK_OPS; ++i) {
    const int idx = tid + i * TPB;
    const int r   = idx / (DDIM / 4);
    const int c4  = idx % (DDIM / 4);
    int g = row0 + r;
    if (g >= rowLimit) g = 0;                 // clamp (keeps ASYNCcnt uniform)
    const float* gp = gbase + (size_t)g * DDIM + c4 * 4;
    float* lp = dst + r * LDS_STRIDE + c4 * 4;
#if HAVE_ASYNC
    __builtin_amdgcn_global_load_async_to_lds_b128(
        (as1_v4i*)(size_t)gp,
        (as3_v4i*)(unsigned int)(size_t)lp,   // low 32 bits of generic = LDS addr
        0, 0);
#else
    const float4 v = *(const float4*)gp;
    lp[0] = v.x; lp[1] = v.y; lp[2] = v.z; lp[3] = v.w;
#endif
  }
}

__global__ __launch_bounds__(TPB)
void attn_wmma_f32(const float* __restrict__ x1,
                   const float* __restrict__ x2,
                   const unsigned char* __restrict__ mask,
                   float* __restrict__ out) {
  __shared__ float sV[2][QT * LDS_STRIDE];  // buf0 doubles as the Q tile
  __shared__ float sS[QT * S_STRIDE];       // scores -> probabilities

  const int q0   = blockIdx.x * QT;
  const int b    = blockIdx.y;
  const int tid  = threadIdx.x;
  const int wave = tid >> 5;
  const int lane = tid & 31;
  const int half = lane >> 4;      // 0: lanes 0-15, 1: lanes 16-31
  const int l16  = lane & 15;

  const float* x1b = x1 + ((size_t)b * QLEN + q0) * DDIM;
  const float* x2b = x2 + (size_t)b * KLEN * DDIM;

  // ---- Phase 0: stage Q tile [16 x 768] into sV[0] (async) --------------
  stage16(x1b, &sV[0][0], 0, QT, tid);
  WAIT_ASYNC_ALL();
  __syncthreads();

  // ---- Phase 1: S = Q * K^T  (wave w covers key cols [w*128, w*128+128))
  {
    const float* sQ = &sV[0][0];
    const int n0 = wave * 128;
    for (int nt = 0; nt < 8; ++nt) {
      const int n    = n0 + nt * 16;
      const int krow = n + l16;              // key row this lane supplies for B
      v8f acc = {0.f, 0.f, 0.f, 0.f, 0.f, 0.f, 0.f, 0.f};
      if (n + 16 <= KLEN) {                  // fast path: no bounds guard
        const float* kp = x2b + (size_t)krow * DDIM;
        for (int d = 0; d < DDIM; d += 4) {
          v2f a;
          const float* qp = &sQ[l16 * LDS_STRIDE + d + 2 * half];
          a.x = qp[0]; a.y = qp[1];          // A: VGPR0=K{0,2}, VGPR1=K{1,3}
          const float2 kv = *(const float2*)(kp + d + 2 * half);
          v2f bb; bb.x = kv.x; bb.y = kv.y;  // B: VGPR0=K{0,2}, VGPR1=K{1,3}
          acc = __builtin_amdgcn_wmma_f32_16x16x4_f32(
              false, a, false, bb, (short)0, acc, false, false);
        }
      } else {                               // boundary tile (key 511 padded)
        const bool kval = (krow < KLEN);
        const float* kp = x2b + (size_t)(kval ? krow : 0) * DDIM;
        for (int d = 0; d < DDIM; d += 4) {
          v2f a;
          const float* qp = &sQ[l16 * LDS_STRIDE + d + 2 * half];
          a.x = qp[0]; a.y = qp[1];
          const float2 kv = *(const float2*)(kp + d + 2 * half);
          v2f bb;
          bb.x = kval ? kv.x : 0.0f;
          bb.y = kval ? kv.y : 0.0f;
          acc = __builtin_amdgcn_wmma_f32_16x16x4_f32(
              false, a, false, bb, (short)0, acc, false, false);
        }
      }
#pragma unroll
      for (int r = 0; r < 8; ++r)            // C: VGPRr = rows r / r+8
        sS[(r + 8 * half) * S_STRIDE + n + l16] = acc[r];
    }
  }
  __syncthreads();   // all waves done reading Q from sV[0]

  // Prefetch V chunk 0 into sV[0]; overlaps with the softmax phase.
  stage16(x2b, &sV[0][0], 0, KLEN, tid);

  // ---- Phase 2: row softmax over k<511, then inverted dropout -----------
  {
    const int row = tid >> 3;                // 16 rows, 8 lanes per row
    const int sub = tid & 7;
    float mx = -INFINITY;
    for (int j = 0; j < 64; ++j) {
      const int c = sub + j * 8;
      if (c < KLEN) mx = fmaxf(mx, sS[row * S_STRIDE + c]);
    }
    mx = fmaxf(mx, __shfl_xor(mx, 4, 8));
    mx = fmaxf(mx, __shfl_xor(mx, 2, 8));
    mx = fmaxf(mx, __shfl_xor(mx, 1, 8));
    float sum = 0.0f;
    for (int j = 0; j < 64; ++j) {
      const int c = sub + j * 8;
      if (c < KLEN) {
        const float e = __expf(sS[row * S_STRIDE + c] - mx);
        sS[row * S_STRIDE + c] = e;
        sum += e;
      }
    }
    sum += __shfl_xor(sum, 4, 8);
    sum += __shfl_xor(sum, 2, 8);
    sum += __shfl_xor(sum, 1, 8);
    const float inv = INV_KEEP / sum;
    const unsigned char* mrow = mask + ((size_t)b * QLEN + q0 + row) * KLEN;
    for (int j = 0; j < 64; ++j) {
      const int c = sub + j * 8;
      if (c < KLEN) {
        const float p = sS[row * S_STRIDE + c] * inv;
        sS[row * S_STRIDE + c] = mrow[c] ? p : 0.0f;
      } else {
        sS[row * S_STRIDE + c] = 0.0f;       // padded key column 511
      }
    }
  }
  __syncthreads();

  // ---- Phase 3: O = P * V, double-buffered async V staging --------------
  {
    const int n0 = wave * 192;               // 12 output n-tiles per wave
    v8f acc[12];
#pragma unroll
    for (int t = 0; t < 12; ++t) acc[t] = (v8f){0.f,0.f,0.f,0.f,0.f,0.f,0.f,0.f};

    for (int kc = 0; kc < 32; ++kc) {
      const int kbase = kc * 16;
      float* cur = &sV[kc & 1][0];
      if (kc + 1 < 32) {                     // prefetch next chunk (other buf)
        stage16(x2b, &sV[(kc + 1) & 1][0], kbase + 16, KLEN, tid);
        WAIT_ASYNC_PREV();                   // older 24 per-wave ops complete
      } else {
        WAIT_ASYNC_ALL();
      }
      __syncthreads();                       // chunk kc visible to all waves
      if (kbase + QT > KLEN) {               // last chunk: zero padded row 511
        for (int c = tid; c < DDIM; c += TPB) cur[(QT - 1) * LDS_STRIDE + c] = 0.0f;
        __syncthreads();
      }
#pragma unroll
      for (int kk = 0; kk < 16; kk += 4) {
        v2f a;
        const float* pp = &sS[l16 * S_STRIDE + kbase + kk + 2 * half];
        a.x = pp[0]; a.y = pp[1];
#pragma unroll
        for (int t = 0; t < 12; ++t) {
          const int n = n0 + t * 16;
          v2f bb;
          bb.x = cur[(kk + 2 * half)     * LDS_STRIDE + n + l16];
          bb.y = cur[(kk + 2 * half + 1) * LDS_STRIDE + n + l16];
          acc[t] = __builtin_amdgcn_wmma_f32_16x16x4_f32(
              false, a, false, bb, (short)0, acc[t], false, false);
        }
      }
      __syncthreads();                       // cur fully consumed before reuse
    }

    float* ob = out + ((size_t)b * QLEN + q0) * DDIM;
#pragma unroll
    for (int t = 0; t < 12; ++t) {
      const int n = n0 + t * 16;
#pragma unroll
      for (int r = 0; r < 8; ++r)
        ob[(size_t)(r + 8 * half) * DDIM + n + l16] = acc[t][r];
    }
  }
}

extern "C" void kernel_launch(void* const* d_in, const int* in_sizes, int n_in,
                              void* d_out, int out_size, void* d_ws, size_t ws_size,
                              hipStream_t stream) {
  const float* x1 = (const float*)d_in[0];                   // [32,512,768] f32
  const float* x2 = (const float*)d_in[1];                   // [32,511,768] f32
  const unsigned char* mask = (const unsigned char*)d_in[2]; // [32,512,511] bool
  float* out = (float*)d_out;                                // [32,512,768] f32

  dim3 grid(QLEN / QT, BATCH);   // (32, 32)
  dim3 block(TPB);               // 128 threads = 4 waves
  attn_wmma_f32<<<grid, block, 0, stream>>>(x1, x2, mask, out);
}